// GTNLiteLayer_30365418783290
// MI455X (gfx1250) — compile-verified
//
#include <hip/hip_runtime.h>
#include <math.h>

typedef __attribute__((ext_vector_type(2))) float v2f;
typedef __attribute__((ext_vector_type(8))) float v8f;
typedef __attribute__((ext_vector_type(4))) unsigned int v4u;
typedef __attribute__((ext_vector_type(8))) int v8i;
typedef __attribute__((ext_vector_type(4))) int v4i;

#define DDIM 128
#define AROW_STRIDE 132   // 128 + 4 DWORD pad (TDM pad_interval=6, pad_amount=3)

// ---------------------------------------------------------------------------
// Zero the aggregation buffer + counts (grid-stride).
// ---------------------------------------------------------------------------
__global__ void gtn_zero_kernel(float* __restrict__ agg, float* __restrict__ cnt,
                                int n_rows) {
    long total  = (long)n_rows * DDIM;
    long i      = (long)blockIdx.x * blockDim.x + threadIdx.x;
    long stride = (long)gridDim.x * blockDim.x;
    for (long j = i; j < total; j += stride) agg[j] = 0.0f;
    for (long j = i; j < n_rows; j += stride) cnt[j] = 0.0f;
}

// ---------------------------------------------------------------------------
// Edge scatter: one wave32 per edge; float4 per lane; native f32 atomics into
// the L2-resident aggregation buffer. Lane 0 counts the edge.
// ---------------------------------------------------------------------------
__global__ __launch_bounds__(256)
void gtn_scatter_kernel(const float* __restrict__ xsrc,
                        const int* __restrict__ src,
                        const int* __restrict__ dst,
                        float* __restrict__ agg,
                        float* __restrict__ cnt,
                        int E) {
    int wave = (int)((blockIdx.x * blockDim.x + threadIdx.x) >> 5);
    int lane = threadIdx.x & 31;
    if (wave >= E) return;
    int s = src[wave];
    int d = dst[wave];
    const float4 v = *(const float4*)(xsrc + (long)s * DDIM + lane * 4);
    float* base = agg + (long)d * DDIM + lane * 4;
    __hip_atomic_fetch_add(base + 0, v.x, __ATOMIC_RELAXED, __HIP_MEMORY_SCOPE_AGENT);
    __hip_atomic_fetch_add(base + 1, v.y, __ATOMIC_RELAXED, __HIP_MEMORY_SCOPE_AGENT);
    __hip_atomic_fetch_add(base + 2, v.z, __ATOMIC_RELAXED, __HIP_MEMORY_SCOPE_AGENT);
    __hip_atomic_fetch_add(base + 3, v.w, __ATOMIC_RELAXED, __HIP_MEMORY_SCOPE_AGENT);
    if (lane == 0)
        __hip_atomic_fetch_add(cnt + d, 1.0f, __ATOMIC_RELAXED, __HIP_MEMORY_SCOPE_AGENT);
}

// ---------------------------------------------------------------------------
// Tensor Data Mover: 2-D tile load Global -> LDS.
// D# group0: {count=1, lds_addr, global_addr[56:0], type=2}
// D# group1: data_size=4B, optional LDS pad (+4 DWORDs / 128 DWORDs),
//            tensor dims (OOB rows read as zero), tile dims, dim0 stride.
// Groups 2/3 zero (tensor <= 2D).
// ---------------------------------------------------------------------------
__device__ __forceinline__ unsigned gtn_lds_off(const void* p) {
    // generic LDS pointer: low 32 bits are the LDS byte offset (flat aperture)
    return (unsigned)(unsigned long long)p;
}

__device__ __forceinline__ void gtn_tdm_load_2d(unsigned lds_off, const void* gptr,
                                                unsigned tensor_d0, unsigned tensor_d1,
                                                unsigned tile_d0, unsigned tile_d1,
                                                unsigned stride0, int pad) {
    unsigned long long ga = (unsigned long long)gptr;
    v4u g0;
    g0.x = 1u;                                            // count=1, user mode
    g0.y = lds_off;                                       // LDS byte address
    g0.z = (unsigned)(ga & 0xFFFFFFFFu);                  // global addr lo
    g0.w = (unsigned)((ga >> 32) & 0x01FFFFFFu) | (2u << 30);  // addr hi | type=2
    v8i g1;
    unsigned w0 = (2u << 16);                             // data_size = 4 bytes
    if (pad) w0 |= (1u << 20) | (6u << 22) | (3u << 25);  // pad 4 DW / 128 DW
    g1[0] = (int)w0;
    g1[1] = (int)((tensor_d0 & 0xFFFFu) << 16);                              // dim0 lo
    g1[2] = (int)(((tensor_d0 >> 16) & 0xFFFFu) | ((tensor_d1 & 0xFFFFu) << 16));
    g1[3] = (int)(((tensor_d1 >> 16) & 0xFFFFu) | ((tile_d0 & 0xFFFFu) << 16));
    g1[4] = (int)(tile_d1 & 0xFFFFu);                     // tile_dim1 | tile_dim2=0
    g1[5] = (int)stride0;                                 // tensor_dim0_stride lo
    g1[6] = 0;                                            // stride0 hi | stride1 lo
    g1[7] = 0;
    v4i z4 = {0, 0, 0, 0};
    v8i z8 = {0, 0, 0, 0, 0, 0, 0, 0};
    // 6-arg toolchain variant: (g0, g1, g2, g3, g4, cpol)
    __builtin_amdgcn_tensor_load_to_lds(g0, g1, z4, z4, z8, 0);
}

// ---------------------------------------------------------------------------
// Fused SAGE transform: out(+)= rw * (mean @ Wl + bl + x_dst @ Wr)
// 256 threads = 8 waves; 128 rows per block. Wave w owns row tile w and all
// 8 column tiles via v_wmma_f32_16x16x4_f32. All operands staged in LDS by
// the Tensor Data Mover: Wl/Wr double-buffered, agg/x tiles row-padded by the
// TDM pad feature (132-float stride -> bank-conflict-free A fragments).
// mean = agg * 1/max(cnt,1) folded into the A-fragment load.
// ---------------------------------------------------------------------------
__global__ __launch_bounds__(256)
void gtn_sage_gemm_kernel(float* __restrict__ out,
                          const float* __restrict__ agg,
                          const float* __restrict__ cnt,
                          const float* __restrict__ xdst,
                          const float* __restrict__ Wl,
                          const float* __restrict__ bl,
                          const float* __restrict__ Wr,
                          const float* __restrict__ rel_logits,
                          int rel, int accum, int N) {
    __shared__ float lds_W0[DDIM * DDIM];         // 64 KB: Wl
    __shared__ float lds_W1[DDIM * DDIM];         // 64 KB: Wr
    __shared__ float lds_A0[DDIM * AROW_STRIDE];  // 66 KB: mean-source (agg) tile
    __shared__ float lds_A1[DDIM * AROW_STRIDE];  // 66 KB: x_dst tile

    const int tid  = threadIdx.x;
    const int wave = tid >> 5;
    const int lane = tid & 31;
    const int half = lane >> 4;   // A: K+{0,1} vs K+{2,3}; B: K rows 0,1 vs 2,3
    const int l    = lane & 15;

    const int row0  = blockIdx.x * 128;   // block row base
    const int wrow0 = row0 + wave * 16;   // this wave's row tile

    // Kick off all four TDM transfers from one wave; TENSORcnt is in-order,
    // so waiting <=2 gates pass 1 and <=0 gates pass 2 (Wr/x fetch overlaps
    // pass-1 compute).
    if (wave == 0) {
        gtn_tdm_load_2d(gtn_lds_off(lds_W0), Wl, DDIM, DDIM, DDIM, DDIM, DDIM, 0);
        gtn_tdm_load_2d(gtn_lds_off(lds_A0), agg + (long)row0 * DDIM,
                        DDIM, (unsigned)(N - row0), DDIM, DDIM, DDIM, 1);
        gtn_tdm_load_2d(gtn_lds_off(lds_W1), Wr, DDIM, DDIM, DDIM, DDIM, DDIM, 0);
        gtn_tdm_load_2d(gtn_lds_off(lds_A1), xdst + (long)row0 * DDIM,
                        DDIM, (unsigned)(N - row0), DDIM, DDIM, DDIM, 1);
        __builtin_amdgcn_s_wait_tensorcnt(2);
    }

    // softmax relation weight (3 floats, L2-resident)
    float r0 = rel_logits[0], r1 = rel_logits[1], r2 = rel_logits[2];
    float m  = fmaxf(r0, fmaxf(r1, r2));
    float e0 = __expf(r0 - m), e1 = __expf(r1 - m), e2 = __expf(r2 - m);
    float rw = ((rel == 0) ? e0 : (rel == 1) ? e1 : e2) / (e0 + e1 + e2);

    const int crow  = wrow0 + l;
    const int crow_c = (crow < N) ? crow : (N - 1);
    float rinv = 1.0f / fmaxf(cnt[crow_c], 1.0f);

    const v8f vzero = {0.f, 0.f, 0.f, 0.f, 0.f, 0.f, 0.f, 0.f};
    v8f acc[8];
#pragma unroll
    for (int t = 0; t < 8; ++t) acc[t] = vzero;

    __syncthreads();  // Wl + agg tile resident

    // ---- pass 1: mean @ Wl  (A = agg_tile * rinv) -------------------------
    {
        const float* aptr = lds_A0 + (wave * 16 + l) * AROW_STRIDE + 2 * half;
        for (int k0 = 0; k0 < DDIM; k0 += 4) {
            float2 av = *(const float2*)(aptr + k0);
            v2f a;
            a.x = av.x * rinv;
            a.y = av.y * rinv;
            const int kb = k0 + 2 * half;
#pragma unroll
            for (int t = 0; t < 8; ++t) {
                v2f b;
                b.x = lds_W0[kb * DDIM + t * 16 + l];
                b.y = lds_W0[(kb + 1) * DDIM + t * 16 + l];
                acc[t] = __builtin_amdgcn_wmma_f32_16x16x4_f32(
                    false, a, false, b, (short)0, acc[t], false, false);
            }
        }
    }

    if (wave == 0) __builtin_amdgcn_s_wait_tensorcnt(0);
    __syncthreads();  // Wr + x tile resident

    // ---- pass 2: x_dst @ Wr (accumulate into same C) ----------------------
    {
        const float* aptr = lds_A1 + (wave * 16 + l) * AROW_STRIDE + 2 * half;
        for (int k0 = 0; k0 < DDIM; k0 += 4) {
            float2 av = *(const float2*)(aptr + k0);
            v2f a;
            a.x = av.x;
            a.y = av.y;
            const int kb = k0 + 2 * half;
#pragma unroll
            for (int t = 0; t < 8; ++t) {
                v2f b;
                b.x = lds_W1[kb * DDIM + t * 16 + l];
                b.y = lds_W1[(kb + 1) * DDIM + t * 16 + l];
                acc[t] = __builtin_amdgcn_wmma_f32_16x16x4_f32(
                    false, a, false, b, (short)0, acc[t], false, false);
            }
        }
    }

    // ---- epilogue: bias, relation weight, (optional) accumulate, store ----
#pragma unroll
    for (int t = 0; t < 8; ++t) {
        const int col  = t * 16 + l;
        const float bb = bl[col];
#pragma unroll
        for (int i = 0; i < 8; ++i) {
            const int row = wrow0 + i + half * 8;  // C/D layout: M split at 8
            if (row < N) {
                float v = rw * (acc[t][i] + bb);
                float* p = out + (long)row * DDIM + col;
                if (accum) v += *p;   // this wave exclusively owns these rows
                *p = v;
            }
        }
    }
}

// ---------------------------------------------------------------------------
// rw = softmax(rel_logits) -> output tail
// ---------------------------------------------------------------------------
__global__ void gtn_rw_kernel(const float* __restrict__ rl, float* __restrict__ out_rw) {
    if (threadIdx.x == 0 && blockIdx.x == 0) {
        float m  = fmaxf(rl[0], fmaxf(rl[1], rl[2]));
        float a  = __expf(rl[0] - m);
        float b  = __expf(rl[1] - m);
        float c  = __expf(rl[2] - m);
        float s  = a + b + c;
        out_rw[0] = a / s;
        out_rw[1] = b / s;
        out_rw[2] = c / s;
    }
}

extern "C" void kernel_launch(void* const* d_in, const int* in_sizes, int n_in,
                              void* d_out, int out_size, void* d_ws, size_t ws_size,
                              hipStream_t stream) {
    const float* x_a  = (const float*)d_in[0];
    const float* x_b  = (const float*)d_in[1];
    const int*   e0s  = (const int*)d_in[2];
    const int*   e0d  = (const int*)d_in[3];
    const int*   e1s  = (const int*)d_in[4];
    const int*   e1d  = (const int*)d_in[5];
    const int*   e2s  = (const int*)d_in[6];
    const int*   e2d  = (const int*)d_in[7];
    const float* Wl0  = (const float*)d_in[8];
    const float* bl0  = (const float*)d_in[9];
    const float* Wr0  = (const float*)d_in[10];
    const float* Wl1  = (const float*)d_in[11];
    const float* bl1  = (const float*)d_in[12];
    const float* Wr1  = (const float*)d_in[13];
    const float* Wl2  = (const float*)d_in[14];
    const float* bl2  = (const float*)d_in[15];
    const float* Wr2  = (const float*)d_in[16];
    const float* rl   = (const float*)d_in[17];

    const int Na = in_sizes[0] / DDIM;
    const int Nb = in_sizes[1] / DDIM;
    const int E  = in_sizes[2];
    const int Nmax = (Na > Nb) ? Na : Nb;

    float* out_a  = (float*)d_out;
    float* out_b  = out_a + (long)Na * DDIM;
    float* out_rw = out_b + (long)Nb * DDIM;

    // workspace: [agg: Nmax*128 f32][cnt: Nmax f32]  (~103 MB)
    float* agg = (float*)d_ws;
    float* cnt = agg + (long)Nmax * DDIM;

    dim3 blk(256);
    const int zgrid  = 2048;                    // grid-stride zero fill
    const int sgrid  = (E + 7) / 8;             // 8 edges (waves) per block
    const int ggridA = (Na + 127) / 128;
    const int ggridB = (Nb + 127) / 128;

    // relation 0: a -> a   (out_a = rw0 * h0)
    gtn_zero_kernel<<<zgrid, blk, 0, stream>>>(agg, cnt, Na);
    gtn_scatter_kernel<<<sgrid, blk, 0, stream>>>(x_a, e0s, e0d, agg, cnt, E);
    gtn_sage_gemm_kernel<<<ggridA, blk, 0, stream>>>(out_a, agg, cnt, x_a,
                                                     Wl0, bl0, Wr0, rl, 0, 0, Na);
    // relation 2: b -> a   (out_a += rw2 * h2)
    gtn_zero_kernel<<<zgrid, blk, 0, stream>>>(agg, cnt, Na);
    gtn_scatter_kernel<<<sgrid, blk, 0, stream>>>(x_b, e2s, e2d, agg, cnt, E);
    gtn_sage_gemm_kernel<<<ggridA, blk, 0, stream>>>(out_a, agg, cnt, x_a,
                                                     Wl2, bl2, Wr2, rl, 2, 1, Na);
    // relation 1: a -> b   (out_b = rw1 * h1)
    gtn_zero_kernel<<<zgrid, blk, 0, stream>>>(agg, cnt, Nb);
    gtn_scatter_kernel<<<sgrid, blk, 0, stream>>>(x_a, e1s, e1d, agg, cnt, E);
    gtn_sage_gemm_kernel<<<ggridB, blk, 0, stream>>>(out_b, agg, cnt, x_b,
                                                     Wl1, bl1, Wr1, rl, 1, 0, Nb);

    gtn_rw_kernel<<<1, 32, 0, stream>>>(rl, out_rw);
}